// PONITA_63694365000070
// MI455X (gfx1250) — compile-verified
//
#include <hip/hip_runtime.h>
#include <math.h>

#define NN      10000            // N_NODES
#define NE      80000            // N_EDGES
#define NO      12               // N_ORI
#define IND     16               // IN_DIM
#define CH      64               // C
#define HID     256              // WID*C
#define PF2V    30
#define REDGE   (NE*NO)          // 960000 (div by 16)
#define RNODE   (NN*NO)          // 120000 (div by 16)
#define NG      16

typedef float v2f __attribute__((ext_vector_type(2)));
typedef float v8f __attribute__((ext_vector_type(8)));

static __device__ __forceinline__ v8f wmma4(v2f a, v2f b, v8f c) {
  // D = A(16x4,f32) * B(4x16,f32) + C(16x16,f32)
  return __builtin_amdgcn_wmma_f32_16x16x4_f32(false, a, false, b, (short)0, c, false, false);
}
static __device__ __forceinline__ float geluf(float x) {
  return 0.5f * x * (1.0f + erff(x * 0.7071067811865475f));
}

// ---------------- kb_ori MLP (144 rows, tiny -> VALU) ----------------
__global__ void k_kbori1(const float* __restrict__ og, const float* __restrict__ Wo1,
                         const float* __restrict__ bo1, float* __restrict__ tmp) {
  int t = blockIdx.x * blockDim.x + threadIdx.x;
  if (t >= 144 * CH) return;
  int r = t / CH, c = t - r * CH;
  int i = r / NO, j = r - i * NO;
  float a = og[i*3]*og[j*3] + og[i*3+1]*og[j*3+1] + og[i*3+2]*og[j*3+2];
  float f1 = a, f2 = a*a, f3 = f2*a, f4 = f3*a;      // PF1 = [a, a^2, a^3, a^4]
  float s = bo1[c] + f1*Wo1[0*CH+c] + f2*Wo1[1*CH+c] + f3*Wo1[2*CH+c] + f4*Wo1[3*CH+c];
  tmp[t] = geluf(s);
}
__global__ void k_kbori2(const float* __restrict__ tmp, const float* __restrict__ Wo2,
                         const float* __restrict__ bo2, float* __restrict__ kbori) {
  int t = blockIdx.x * blockDim.x + threadIdx.x;
  if (t >= 144 * CH) return;
  int r = t / CH, c = t - r * CH;
  float s = bo2[c];
  for (int b = 0; b < CH; ++b) s += tmp[r*CH+b] * Wo2[b*CH+c];
  kbori[t] = geluf(s);
}
__global__ void k_fk(const float* __restrict__ kbori, const float* __restrict__ Wf,
                     float* __restrict__ fk) {
  int t = blockIdx.x * blockDim.x + threadIdx.x;
  if (t >= 2 * 144 * CH) return;
  int l = t / (144*CH); int rc = t - l*144*CH; int r = rc / CH, c = rc - r*CH;
  float s = 0.f;
  for (int b = 0; b < CH; ++b) s += kbori[r*CH+b] * Wf[l*CH*CH + b*CH + c];
  fk[t] = s;
}

// ---------------- node embedding x0 = f @ We (WMMA) ----------------
__global__ __launch_bounds__(128) void k_embed(const float* __restrict__ f,
                                               const float* __restrict__ We,
                                               float* __restrict__ x0) {
  const int wid = threadIdx.x >> 5, lane = threadIdx.x & 31;
  const int half = lane >> 4, nlan = lane & 15, koff = half * 2;
  int tile = blockIdx.x * 4 + wid;
  if (tile >= NN / 16) return;
  int base = tile * 16;
  for (int nt = 0; nt < 4; ++nt) {
    v8f acc; for (int i = 0; i < 8; ++i) acc[i] = 0.f;
    for (int kblk = 0; kblk < 4; ++kblk) {
      int k = kblk * 4 + koff;
      v2f a; a.x = f[(size_t)(base+nlan)*IND + k]; a.y = f[(size_t)(base+nlan)*IND + k + 1];
      v2f b; b.x = We[k*CH + nt*16 + nlan];        b.y = We[(k+1)*CH + nt*16 + nlan];
      acc = wmma4(a, b, acc);
    }
    for (int v = 0; v < 8; ++v) {
      int m = v + 8*half;
      x0[(size_t)(base+m)*CH + nt*16 + nlan] = acc[v];
    }
  }
}

// ---------------- per-edge kb: geometry + poly feats + 2-layer WMMA MLP ----------------
__global__ __launch_bounds__(128) void k_kb(const float* __restrict__ pos,
                                            const float* __restrict__ og,
                                            const int* __restrict__ ei,
                                            const float* __restrict__ Wb1, const float* __restrict__ bb1,
                                            const float* __restrict__ Wb2, const float* __restrict__ bb2,
                                            float* __restrict__ kb) {
  __shared__ float feat[4][16][33];
  __shared__ float hid[4][16][68];
  __shared__ float cut[4][16];
  const int wid = threadIdx.x >> 5, lane = threadIdx.x & 31;
  const int half = lane >> 4, nlan = lane & 15, koff = half * 2;
  const int tile = blockIdx.x * 4 + wid;        // grid sized exactly, no OOB
  const int base = tile * 16;

  if (lane < 16) {
    int r = base + lane;
    int e = r / NO, o = r - e * NO;
    int s = ei[e], d = ei[NE + e];
    float rx = pos[s*3+0]-pos[d*3+0], ry = pos[s*3+1]-pos[d*3+1], rz = pos[s*3+2]-pos[d*3+2];
    float dist = sqrtf(rx*rx + ry*ry + rz*rz);
    float ox = og[o*3+0], oy = og[o*3+1], oz = og[o*3+2];
    float a1 = rx*ox + ry*oy + rz*oz;
    float tx = rx - a1*ox, ty = ry - a1*oy, tz = rz - a1*oz;
    float a2 = sqrtf(tx*tx + ty*ty + tz*tz);
    // poly_features([a1,a2], degree 3) -> 2+4+8+16 = 30
    float curv[16]; curv[0] = a1; curv[1] = a2;
    feat[wid][lane][0] = a1; feat[wid][lane][1] = a2;
    int curlen = 2, idx = 2;
    for (int dd = 0; dd < 3; ++dd) {
      float nxt[16]; int nl = 0;
      for (int i = 0; i < curlen; ++i) { nxt[nl++] = curv[i]*a1; nxt[nl++] = curv[i]*a2; }
      for (int i = 0; i < nl; ++i) { curv[i] = nxt[i]; feat[wid][lane][idx+i] = nxt[i]; }
      idx += nl; curlen = nl;
    }
    feat[wid][lane][30] = 0.f; feat[wid][lane][31] = 0.f;
    float u = dist, u2 = u*u, u3 = u2*u, u6 = u3*u3, u7 = u6*u, u8 = u7*u;
    float env = 1.0f - 28.0f*u6 + 48.0f*u7 - 21.0f*u8;   // p=6 polynomial cutoff
    cut[wid][lane] = (dist < 1.0f) ? env : 0.0f;
  }
  __syncthreads();

  // layer 1: feat(16x32, valid K=30) @ Wb1(30x64) + bb1 -> gelu -> hid
  for (int nt = 0; nt < 4; ++nt) {
    float bias = bb1[nt*16 + nlan];
    v8f acc; for (int i = 0; i < 8; ++i) acc[i] = bias;
    for (int kblk = 0; kblk < 8; ++kblk) {
      int k = kblk * 4 + koff;
      v2f a; a.x = feat[wid][nlan][k]; a.y = feat[wid][nlan][k+1];
      v2f b;
      b.x = (k   < PF2V) ? Wb1[k*CH     + nt*16 + nlan] : 0.f;
      b.y = (k+1 < PF2V) ? Wb1[(k+1)*CH + nt*16 + nlan] : 0.f;
      acc = wmma4(a, b, acc);
    }
    for (int v = 0; v < 8; ++v) hid[wid][v + 8*half][nt*16 + nlan] = geluf(acc[v]);
  }
  __syncthreads();

  // layer 2: hid(16x64) @ Wb2(64x64) + bb2 -> gelu * cutoff -> kb
  for (int nt = 0; nt < 4; ++nt) {
    float bias = bb2[nt*16 + nlan];
    v8f acc; for (int i = 0; i < 8; ++i) acc[i] = bias;
    for (int kblk = 0; kblk < 16; ++kblk) {
      int k = kblk * 4 + koff;
      v2f a; a.x = hid[wid][nlan][k];          a.y = hid[wid][nlan][k+1];
      v2f b; b.x = Wb2[k*CH + nt*16 + nlan];   b.y = Wb2[(k+1)*CH + nt*16 + nlan];
      acc = wmma4(a, b, acc);
    }
    for (int v = 0; v < 8; ++v) {
      int m = v + 8*half;
      kb[(size_t)(base+m)*CH + nt*16 + nlan] = geluf(acc[v]) * cut[wid][m];
    }
  }
}

// ---------------- message: (kb @ Wk[l]) * x[src] scattered into x1 ----------------
__global__ __launch_bounds__(128) void k_msg(const float* __restrict__ kb,
                                             const float* __restrict__ Wk_l,
                                             const float* __restrict__ xin,
                                             const int* __restrict__ ei,
                                             float* __restrict__ x1, int layer0) {
  const int wid = threadIdx.x >> 5, lane = threadIdx.x & 31;
  const int half = lane >> 4, nlan = lane & 15, koff = half * 2;
  const int base = (blockIdx.x * 4 + wid) * 16;
  int sA[8], dA[8], oA[8];
  for (int v = 0; v < 8; ++v) {
    int r = base + v + 8*half;
    int e = r / NO, o = r - e * NO;
    sA[v] = ei[e]; dA[v] = ei[NE + e]; oA[v] = o;
  }
  for (int nt = 0; nt < 4; ++nt) {
    v8f acc; for (int i = 0; i < 8; ++i) acc[i] = 0.f;
    for (int kblk = 0; kblk < 16; ++kblk) {
      int k = kblk * 4 + koff;
      v2f a; a.x = kb[(size_t)(base+nlan)*CH + k]; a.y = kb[(size_t)(base+nlan)*CH + k + 1];
      v2f b; b.x = Wk_l[k*CH + nt*16 + nlan];      b.y = Wk_l[(k+1)*CH + nt*16 + nlan];
      acc = wmma4(a, b, acc);
    }
    int col = nt*16 + nlan;
    for (int v = 0; v < 8; ++v) {
      float xv = layer0 ? xin[(size_t)sA[v]*CH + col]
                        : xin[((size_t)sA[v]*NO + oA[v])*CH + col];
      atomicAdd(&x1[((size_t)dA[v]*NO + oA[v])*CH + col], acc[v] * xv);
    }
  }
}

// ---------------- conv einsum + LayerNorm (wave per (v,p) row) ----------------
__global__ __launch_bounds__(128) void k_conv(const float* __restrict__ x1,
                                              const float* __restrict__ fk_l,
                                              const float* __restrict__ cb_l,
                                              const float* __restrict__ g_l,
                                              const float* __restrict__ bl_l,
                                              float* __restrict__ h) {
  const int wid = threadIdx.x >> 5, lane = threadIdx.x & 31;
  int row = blockIdx.x * 4 + wid;                 // (v,p), exact
  int v = row / NO, p = row - v * NO;
  float hc0 = 0.f, hc1 = 0.f;
  int c0 = lane, c1 = lane + 32;
  for (int o = 0; o < NO; ++o) {
    const float* xr = &x1[((size_t)v*NO + o)*CH];
    const float* fr = &fk_l[(o*NO + p)*CH];
    hc0 += xr[c0] * fr[c0];
    hc1 += xr[c1] * fr[c1];
  }
  hc0 = hc0 * (1.0f/12.0f) + cb_l[c0];
  hc1 = hc1 * (1.0f/12.0f) + cb_l[c1];
  float t = hc0 + hc1;
  for (int off = 16; off; off >>= 1) t += __shfl_xor(t, off, 32);
  float mu = t * (1.0f/64.0f);
  float d0 = hc0 - mu, d1 = hc1 - mu;
  float vs = d0*d0 + d1*d1;
  for (int off = 16; off; off >>= 1) vs += __shfl_xor(vs, off, 32);
  float inv = rsqrtf(vs * (1.0f/64.0f) + 1e-5f);
  h[(size_t)row*CH + c0] = d0 * inv * g_l[c0] + bl_l[c0];
  h[(size_t)row*CH + c1] = d1 * inv * g_l[c1] + bl_l[c1];
}

// ---------------- node MLP 64->256(gelu)->64 (+residual) via WMMA ----------------
__global__ __launch_bounds__(128) void k_mlp(const float* __restrict__ h,
                                             const float* __restrict__ W1_l, const float* __restrict__ b1_l,
                                             const float* __restrict__ W2_l, const float* __restrict__ b2_l,
                                             float* __restrict__ x, int residual) {
  __shared__ float chunk[4][16][17];
  const int wid = threadIdx.x >> 5, lane = threadIdx.x & 31;
  const int half = lane >> 4, nlan = lane & 15, koff = half * 2;
  const int base = (blockIdx.x * 4 + wid) * 16;   // exact
  v8f out[4];
  for (int nt = 0; nt < 4; ++nt) {
    float bias = b2_l[nt*16 + nlan];
    for (int i = 0; i < 8; ++i) out[nt][i] = bias;
  }
  for (int nt1 = 0; nt1 < 16; ++nt1) {
    float bias = b1_l[nt1*16 + nlan];
    v8f acc; for (int i = 0; i < 8; ++i) acc[i] = bias;
    for (int kblk = 0; kblk < 16; ++kblk) {
      int k = kblk * 4 + koff;
      v2f a; a.x = h[(size_t)(base+nlan)*CH + k];    a.y = h[(size_t)(base+nlan)*CH + k + 1];
      v2f b; b.x = W1_l[k*HID + nt1*16 + nlan];      b.y = W1_l[(k+1)*HID + nt1*16 + nlan];
      acc = wmma4(a, b, acc);
    }
    for (int v = 0; v < 8; ++v) chunk[wid][v + 8*half][nlan] = geluf(acc[v]);
    __builtin_amdgcn_wave_barrier();   // in-wave LDS store->load ordering (DS is in-order)
    for (int kblk = 0; kblk < 4; ++kblk) {
      int kl = kblk * 4 + koff;
      v2f a2; a2.x = chunk[wid][nlan][kl]; a2.y = chunk[wid][nlan][kl+1];
      int kg = nt1*16 + kl;
      for (int nt2 = 0; nt2 < 4; ++nt2) {
        v2f b2; b2.x = W2_l[kg*CH + nt2*16 + nlan]; b2.y = W2_l[(kg+1)*CH + nt2*16 + nlan];
        out[nt2] = wmma4(a2, b2, out[nt2]);
      }
    }
    __builtin_amdgcn_wave_barrier();
  }
  for (int nt = 0; nt < 4; ++nt) {
    int col = nt*16 + nlan;
    for (int v = 0; v < 8; ++v) {
      size_t off = (size_t)(base + v + 8*half)*CH + col;
      float val = out[nt][v];
      if (residual) val += x[off];
      x[off] = val;
    }
  }
}

// ---------------- readout accumulation ----------------
__global__ void k_readout(const float* __restrict__ x, const float* __restrict__ Wr_l,
                          const float* __restrict__ br_l, float* __restrict__ racc) {
  int r = blockIdx.x * blockDim.x + threadIdx.x;
  if (r >= RNODE) return;
  float s0 = br_l[0], s1 = br_l[1];
  for (int c = 0; c < CH; ++c) {
    float xv = x[(size_t)r*CH + c];
    s0 += xv * Wr_l[c*2 + 0];
    s1 += xv * Wr_l[c*2 + 1];
  }
  racc[r*2 + 0] += 0.5f * s0;   // mean over 2 layers
  racc[r*2 + 1] += 0.5f * s1;
}

// ---------------- final segment sums ----------------
__global__ void k_final(const float* __restrict__ racc, const float* __restrict__ og,
                        const int* __restrict__ batch, float* __restrict__ out) {
  int v = blockIdx.x * blockDim.x + threadIdx.x;
  if (v >= NN) return;
  float s = 0.f, vx = 0.f, vy = 0.f, vz = 0.f;
  for (int o = 0; o < NO; ++o) {
    float r0 = racc[((size_t)v*NO + o)*2 + 0];
    float r1 = racc[((size_t)v*NO + o)*2 + 1];
    s  += r0;
    vx += r1 * og[o*3+0]; vy += r1 * og[o*3+1]; vz += r1 * og[o*3+2];
  }
  const float inv12 = 1.0f / 12.0f;
  int g = batch[v];
  atomicAdd(&out[g],                s  * inv12);
  atomicAdd(&out[NG + g*3 + 0],     vx * inv12);
  atomicAdd(&out[NG + g*3 + 1],     vy * inv12);
  atomicAdd(&out[NG + g*3 + 2],     vz * inv12);
}

extern "C" void kernel_launch(void* const* d_in, const int* in_sizes, int n_in,
                              void* d_out, int out_size, void* d_ws, size_t ws_size,
                              hipStream_t stream) {
  (void)in_sizes; (void)n_in; (void)ws_size;
  const float* pos  = (const float*)d_in[0];
  const float* f    = (const float*)d_in[1];
  const float* og   = (const float*)d_in[2];
  const int*   ei   = (const int*)  d_in[3];
  const int*   bat  = (const int*)  d_in[4];
  const float* Wb1  = (const float*)d_in[5];
  const float* bb1  = (const float*)d_in[6];
  const float* Wb2  = (const float*)d_in[7];
  const float* bb2  = (const float*)d_in[8];
  const float* Wo1  = (const float*)d_in[9];
  const float* bo1  = (const float*)d_in[10];
  const float* Wo2  = (const float*)d_in[11];
  const float* bo2  = (const float*)d_in[12];
  const float* We   = (const float*)d_in[13];
  const float* Wk   = (const float*)d_in[14];
  const float* Wf   = (const float*)d_in[15];
  const float* cb   = (const float*)d_in[16];
  const float* lng  = (const float*)d_in[17];
  const float* lnb  = (const float*)d_in[18];
  const float* W1   = (const float*)d_in[19];
  const float* b1   = (const float*)d_in[20];
  const float* W2   = (const float*)d_in[21];
  const float* b2   = (const float*)d_in[22];
  const float* Wr   = (const float*)d_in[23];
  const float* br   = (const float*)d_in[24];

  char* ws = (char*)d_ws;
  size_t off = 0;
  auto carve = [&](size_t bytes) -> char* {
    char* p = ws + off;
    off = (off + bytes + 255) & ~(size_t)255;
    return p;
  };
  float* kb    = (float*)carve((size_t)REDGE * CH * 4);   // 245.8 MB
  float* x0    = (float*)carve((size_t)NN * CH * 4);
  float* xcur  = (float*)carve((size_t)RNODE * CH * 4);
  float* x1    = (float*)carve((size_t)RNODE * CH * 4);
  float* h     = (float*)carve((size_t)RNODE * CH * 4);
  float* racc  = (float*)carve((size_t)RNODE * 2 * 4);
  float* tmpo  = (float*)carve(144 * CH * 4);
  float* kbori = (float*)carve(144 * CH * 4);
  float* fk    = (float*)carve(2 * 144 * CH * 4);

  hipMemsetAsync(racc, 0, (size_t)RNODE * 2 * 4, stream);
  hipMemsetAsync(d_out, 0, (size_t)out_size * sizeof(float), stream);

  k_kbori1<<<(144*CH + 255)/256, 256, 0, stream>>>(og, Wo1, bo1, tmpo);
  k_kbori2<<<(144*CH + 255)/256, 256, 0, stream>>>(tmpo, Wo2, bo2, kbori);
  k_fk<<<(2*144*CH + 255)/256, 256, 0, stream>>>(kbori, Wf, fk);
  k_embed<<<(NN/16 + 3)/4, 128, 0, stream>>>(f, We, x0);
  k_kb<<<REDGE/16/4, 128, 0, stream>>>(pos, og, ei, Wb1, bb1, Wb2, bb2, kb);

  for (int l = 0; l < 2; ++l) {
    hipMemsetAsync(x1, 0, (size_t)RNODE * CH * 4, stream);
    k_msg<<<REDGE/16/4, 128, 0, stream>>>(kb, Wk + (size_t)l*CH*CH,
                                          (l == 0) ? x0 : xcur, ei, x1, (l == 0) ? 1 : 0);
    k_conv<<<RNODE/4, 128, 0, stream>>>(x1, fk + (size_t)l*144*CH, cb + l*CH,
                                        lng + l*CH, lnb + l*CH, h);
    k_mlp<<<RNODE/16/4, 128, 0, stream>>>(h, W1 + (size_t)l*CH*HID, b1 + l*HID,
                                          W2 + (size_t)l*HID*CH, b2 + l*CH,
                                          xcur, (l == 1) ? 1 : 0);
    k_readout<<<(RNODE + 255)/256, 256, 0, stream>>>(xcur, Wr + l*CH*2, br + l*2, racc);
  }
  k_final<<<(NN + 255)/256, 256, 0, stream>>>(racc, og, bat, (float*)d_out);
}